// DiscriminativeReconstructionLoss_64656437674052
// MI455X (gfx1250) — compile-verified
//
#include <hip/hip_runtime.h>
#include <stdint.h>

#define NBINS 2048
#define SBLK  512
#define TPB   256

typedef __attribute__((ext_vector_type(2))) float v2f;
typedef __attribute__((ext_vector_type(8))) float v8f;

struct __align__(16) Header {
  double T1, T2;        // exact sums of positives / squares
  double SC1, SC2;      // fixed-point scales for deterministic i64 atomics
  float  kf, lo, scale, vmin, vmax;
  unsigned k;
  int pad[2];
};

struct __align__(16) Partial {
  double s1, s2;
  float  mn, mx;
  unsigned cnt, pad;
};

// ---------------- Pass 1: exact stats (deterministic tree reduction) --------
__global__ void drl_stats(const float4* __restrict__ x4,
                          const uchar4* __restrict__ m4,
                          Partial* __restrict__ parts, long long n4) {
  const int tid = threadIdx.x;
  long long gid    = (long long)blockIdx.x * blockDim.x + tid;
  long long stride = (long long)gridDim.x * blockDim.x;
  double s1 = 0.0, s2 = 0.0;
  unsigned cnt = 0;
  float mn = 3e38f, mx = -3e38f;
  for (long long i = gid; i < n4; i += stride) {
    if (i + 2 * stride < n4) __builtin_prefetch((const void*)&x4[i + 2 * stride], 0, 1);
    float4 v  = x4[i];
    uchar4 mk = m4[i];
    if (mk.x) { cnt++; s1 += v.x; s2 += (double)v.x * v.x; mn = fminf(mn, v.x); mx = fmaxf(mx, v.x); }
    if (mk.y) { cnt++; s1 += v.y; s2 += (double)v.y * v.y; mn = fminf(mn, v.y); mx = fmaxf(mx, v.y); }
    if (mk.z) { cnt++; s1 += v.z; s2 += (double)v.z * v.z; mn = fminf(mn, v.z); mx = fmaxf(mx, v.z); }
    if (mk.w) { cnt++; s1 += v.w; s2 += (double)v.w * v.w; mn = fminf(mn, v.w); mx = fmaxf(mx, v.w); }
  }
  __shared__ double  rs1[TPB], rs2[TPB];
  __shared__ float   rmn[TPB], rmx[TPB];
  __shared__ unsigned rc[TPB];
  rs1[tid] = s1; rs2[tid] = s2; rmn[tid] = mn; rmx[tid] = mx; rc[tid] = cnt;
  __syncthreads();
  for (int s = TPB / 2; s > 0; s >>= 1) {
    if (tid < s) {
      rs1[tid] += rs1[tid + s]; rs2[tid] += rs2[tid + s];
      rmn[tid] = fminf(rmn[tid], rmn[tid + s]);
      rmx[tid] = fmaxf(rmx[tid], rmx[tid + s]);
      rc[tid]  += rc[tid + s];
    }
    __syncthreads();
  }
  if (tid == 0) {
    Partial p; p.s1 = rs1[0]; p.s2 = rs2[0]; p.mn = rmn[0]; p.mx = rmx[0];
    p.cnt = rc[0]; p.pad = 0;
    parts[blockIdx.x] = p;
  }
}

// ---------------- Pass 1b: finalize header + zero bins ----------------------
__global__ void drl_finalize(const Partial* __restrict__ parts, Header* __restrict__ hdr,
                             unsigned* __restrict__ bc,
                             unsigned long long* __restrict__ b1,
                             unsigned long long* __restrict__ b2, int np) {
  __shared__ double  ls1[512], ls2[512];
  __shared__ float   lmn[512], lmx[512];
  __shared__ unsigned lc[512];
  const int tid = threadIdx.x;
  double s1 = 0.0, s2 = 0.0; float mn = 3e38f, mx = -3e38f; unsigned c = 0;
  if (tid < np) { Partial p = parts[tid]; s1 = p.s1; s2 = p.s2; mn = p.mn; mx = p.mx; c = p.cnt; }
  ls1[tid] = s1; ls2[tid] = s2; lmn[tid] = mn; lmx[tid] = mx; lc[tid] = c;
  __syncthreads();
  for (int s = 256; s > 0; s >>= 1) {
    if (tid < s) {
      ls1[tid] += ls1[tid + s]; ls2[tid] += ls2[tid + s];
      lmn[tid] = fminf(lmn[tid], lmn[tid + s]);
      lmx[tid] = fmaxf(lmx[tid], lmx[tid + s]);
      lc[tid]  += lc[tid + s];
    }
    __syncthreads();
  }
  if (tid == 0) {
    Header h;
    unsigned k = lc[0];
    h.T1 = ls1[0]; h.T2 = ls2[0];
    h.vmin = lmn[0]; h.vmax = lmx[0]; h.lo = lmn[0];
    double amax = fmax(fabs((double)h.vmin), fabs((double)h.vmax));
    if (amax < 1e-30) amax = 1e-30;
    h.SC1 = 0x1p61 / (amax * ((double)k + 1.0));
    h.SC2 = 0x1p61 / (amax * amax * ((double)k + 1.0));
    h.k = k; h.kf = (float)k;
    float range = h.vmax - h.vmin;
    if (!(range > 0.f)) range = 1.f;
    h.scale = ((float)NBINS / range) * (1.0f - 1e-6f);
    h.pad[0] = h.pad[1] = 0;
    *hdr = h;
  }
  for (int b = tid; b < NBINS; b += blockDim.x) { bc[b] = 0; b1[b] = 0; b2[b] = 0; }
}

// ---------------- Pass 2: histogram with async global->LDS staging ----------
__global__ void drl_hist(const float* __restrict__ x,
                         const uint8_t* __restrict__ m,
                         const Header* __restrict__ hdr,
                         unsigned* __restrict__ bc,
                         unsigned long long* __restrict__ b1,
                         unsigned long long* __restrict__ b2, long long n) {
  __shared__ unsigned            lcnt[NBINS];
  __shared__ unsigned long long  ls1[NBINS];
  __shared__ unsigned long long  ls2[NBINS];
  __shared__ __align__(16) float stage[2][TPB * 4];
  const int tid = threadIdx.x;
  for (int b = tid; b < NBINS; b += TPB) { lcnt[b] = 0; ls1[b] = 0; ls2[b] = 0; }
  const float  lo  = hdr->lo, sc = hdr->scale;
  const double SC1 = hdr->SC1, SC2 = hdr->SC2;
  __syncthreads();

  const long long base   = (long long)blockIdx.x * TPB * 4;
  const long long stride = (long long)gridDim.x * TPB * 4;

  // Per-wave-private double-buffered async staging: each lane async-copies its
  // own 16B into its own LDS slot; no cross-wave sharing => no barriers needed.
  auto issue = [&](int buf, long long p) {
    long long idx = p + (long long)tid * 4;
    if (idx < n) {
      unsigned lds = (unsigned)(uintptr_t)&stage[buf][tid * 4];
      unsigned long long gp = (unsigned long long)(uintptr_t)(x + idx);
      asm volatile("global_load_async_to_lds_b128 %0, %1, off"
                   :: "v"(lds), "v"(gp) : "memory");
    }
  };

  auto acc = [&](float xv, unsigned char mk) {
    if (mk) {
      int b = (int)((xv - lo) * sc);
      b = min(max(b, 0), NBINS - 1);
      atomicAdd(&lcnt[b], 1u);
      long long f1 = __double2ll_rn((double)xv * SC1);
      long long f2 = __double2ll_rn((double)xv * (double)xv * SC2);
      atomicAdd(&ls1[b], (unsigned long long)f1);
      atomicAdd(&ls2[b], (unsigned long long)f2);
    }
  };

  int cur = 0;
  issue(0, base);
  for (long long p = base; p < n; p += stride) {
    long long nxt = p + stride;
    if (nxt < n) {
      issue(cur ^ 1, nxt);
      asm volatile("s_wait_asynccnt 0x1" ::: "memory");  // buffer `cur` done
    } else {
      asm volatile("s_wait_asynccnt 0x0" ::: "memory");
    }
    long long idx = p + (long long)tid * 4;
    if (idx < n) {
      float4 v  = *(const float4*)&stage[cur][tid * 4];
      uchar4 mk = *(const uchar4*)(m + idx);
      acc(v.x, mk.x); acc(v.y, mk.y); acc(v.z, mk.z); acc(v.w, mk.w);
    }
    cur ^= 1;
  }
  __syncthreads();
  for (int b = tid; b < NBINS; b += TPB) {
    unsigned c = lcnt[b];
    if (c) {
      atomicAdd(&bc[b], c);
      atomicAdd(&b1[b], ls1[b]);   // i64 fixed-point: deterministic
      atomicAdd(&b2[b], ls2[b]);
    }
  }
}

// ---------------- WMMA 256-element inclusive scan (one wave) ----------------
// S = L * X with L = 16x16 lower-triangular ones (exact in f32), chained as
// four v_wmma_f32_16x16x4_f32, then shfl fix-up of cross-column offsets.
__device__ inline void wave_scan_256(float* v, int lane) {
  const int col = lane & 15, half = lane >> 4;
  v8f d = {};
#pragma unroll
  for (int c = 0; c < 4; ++c) {
    const int k0 = 4 * c + 2 * half;
    v2f a, b;
    a.x = (k0     <= col) ? 1.0f : 0.0f;   // A row M == lane&15
    a.y = (k0 + 1 <= col) ? 1.0f : 0.0f;
    b.x = v[col * 16 + k0];
    b.y = v[col * 16 + k0 + 1];
    d = __builtin_amdgcn_wmma_f32_16x16x4_f32(false, a, false, b, (short)0, d,
                                              false, false);
  }
  float tot = __shfl(d[7], 16 + col, 32);  // column totals S[15][col]
  float p = tot;
#pragma unroll
  for (int s = 1; s < 16; s <<= 1) {
    int src = (lane & 16) | ((col - s) & 15);
    float u = __shfl(p, src, 32);
    if (col >= s) p += u;
  }
  float off = p - tot;                     // exclusive prefix of column totals
#pragma unroll
  for (int r = 0; r < 8; ++r)
    v[col * 16 + r + 8 * half] = d[r] + off;
}

// ---------------- Pass 3: scan bins + Otsu argmin + emit scalar -------------
__global__ void drl_solve(const Header* __restrict__ hdr,
                          const unsigned* __restrict__ bc,
                          const unsigned long long* __restrict__ b1,
                          const unsigned long long* __restrict__ b2,
                          float* __restrict__ out) {
  __shared__ float cs_c[NBINS], cs_1[NBINS], cs_2[NBINS];
  __shared__ float tile_off[3][NBINS / 256];
  __shared__ float red_f[256], red_i[256], red_s[256];
  const int tid = threadIdx.x;
  const double iSC1 = 1.0 / hdr->SC1, iSC2 = 1.0 / hdr->SC2;
  for (int b = tid; b < NBINS; b += 256) {
    cs_c[b] = (float)bc[b];
    cs_1[b] = (float)((double)(long long)b1[b] * iSC1);
    cs_2[b] = (float)((double)(long long)b2[b] * iSC2);
  }
  __syncthreads();

  const int wave = tid >> 5, lane = tid & 31;
  const int NT = NBINS / 256;              // 8 tiles per quantity
  for (int task = wave; task < 3 * NT; task += 8) {
    int q = task / NT, t = task % NT;
    float* arr = (q == 0) ? cs_c : (q == 1) ? cs_1 : cs_2;
    wave_scan_256(arr + t * 256, lane);    // uniform per wave => EXEC all 1s
  }
  __syncthreads();
  if (tid < 3) {
    float* arr = (tid == 0) ? cs_c : (tid == 1) ? cs_1 : cs_2;
    float run = 0.f;
    for (int t = 0; t < NT; ++t) { tile_off[tid][t] = run; run += arr[t * 256 + 255]; }
  }
  __syncthreads();
  for (int b = tid; b < NBINS; b += 256) {
    int t = b >> 8;
    cs_c[b] += tile_off[0][t]; cs_1[b] += tile_off[1][t]; cs_2[b] += tile_off[2][t];
  }
  __syncthreads();

  const float kf = hdr->kf;
  const float T1 = (float)hdr->T1, T2 = (float)hdr->T2;
  float bestf = 3e38f, besti = 3e38f, bests = 0.f;
  for (int b = tid; b < NBINS; b += 256) {
    float i0  = b ? cs_c[b - 1] : 0.f;
    float S10 = b ? cs_1[b - 1] : 0.f;
    float S20 = b ? cs_2[b - 1] : 0.f;
    float c   = cs_c[b] - i0;
    if (c < 0.5f) continue;
    float m1v = (cs_1[b] - S10) / c;
    float m2v = (cs_2[b] - S20) / c;
    float jhi = fminf(c, (kf - 1.f) - i0);
    if (jhi < 1.f) continue;
    auto fj = [&](float j) {
      float i  = i0 + j;
      float a1 = S10 + j * m1v, a2 = S20 + j * m2v;
      float posv = a2 - a1 * a1 / i;
      float r1 = T1 - a1, r2 = T2 - a2;
      float negv = r2 - r1 * r1 / (kf - i);
      return posv + negv;
    };
    float lo_ = 1.f, hi_ = jhi;
    for (int it = 0; it < 64 && hi_ - lo_ > 2.f; ++it) {   // within-bin ternary
      float d3 = (hi_ - lo_) * (1.f / 3.f);
      float ja = floorf(lo_ + d3), jb = ceilf(hi_ - d3);
      if (fj(ja) < fj(jb)) hi_ = jb - 1.f; else lo_ = ja + 1.f;
    }
    for (float j = lo_; j <= hi_; j += 1.f) {
      float fv = fj(j), iv = i0 + j;
      if (fv < bestf || (fv == bestf && iv < besti)) {
        bestf = fv; besti = iv; bests = S10 + j * m1v;
      }
    }
    { // explicit endpoints
      float fv = fj(1.f);
      if (fv < bestf || (fv == bestf && i0 + 1.f < besti)) { bestf = fv; besti = i0 + 1.f; bests = S10 + m1v; }
      fv = fj(jhi);
      if (fv < bestf || (fv == bestf && i0 + jhi < besti)) { bestf = fv; besti = i0 + jhi; bests = S10 + jhi * m1v; }
    }
  }
  red_f[tid] = bestf; red_i[tid] = besti; red_s[tid] = bests;
  __syncthreads();
  for (int s = 128; s > 0; s >>= 1) {
    if (tid < s) {
      float f2 = red_f[tid + s], i2 = red_i[tid + s];
      if (f2 < red_f[tid] || (f2 == red_f[tid] && i2 < red_i[tid])) {
        red_f[tid] = f2; red_i[tid] = i2; red_s[tid] = red_s[tid + s];
      }
    }
    __syncthreads();
  }
  if (tid == 0) {
    float var_tot = (T2 - T1 * T1 / kf) / kf;
    float reg = red_f[0] / var_tot / kf;
    float pos_mean = red_s[0] / red_i[0];
    out[0] = pos_mean + 0.5f * reg;
  }
}

// ---------------------------------------------------------------------------
extern "C" void kernel_launch(void* const* d_in, const int* in_sizes, int n_in,
                              void* d_out, int out_size, void* d_ws, size_t ws_size,
                              hipStream_t stream) {
  const float*   x    = (const float*)d_in[0];
  const uint8_t* mask = (const uint8_t*)d_in[1];
  const long long n = (long long)in_sizes[0];

  Header*  hdr   = (Header*)d_ws;
  Partial* parts = (Partial*)((char*)d_ws + 256);
  char* p = (char*)d_ws + 256 + (size_t)SBLK * sizeof(Partial);
  unsigned*           bin_cnt = (unsigned*)p;            p += (size_t)NBINS * 4;
  unsigned long long* bin_s1  = (unsigned long long*)p;  p += (size_t)NBINS * 8;
  unsigned long long* bin_s2  = (unsigned long long*)p;  // total ws use ~57KB

  drl_stats   <<<SBLK, TPB, 0, stream>>>((const float4*)x, (const uchar4*)mask,
                                         parts, n / 4);
  drl_finalize<<<1, 512, 0, stream>>>(parts, hdr, bin_cnt, bin_s1, bin_s2, SBLK);
  drl_hist    <<<SBLK, TPB, 0, stream>>>(x, mask, hdr, bin_cnt, bin_s1, bin_s2, n);
  drl_solve   <<<1, 256, 0, stream>>>(hdr, bin_cnt, bin_s1, bin_s2, (float*)d_out);
}